// UnrolledResourceAllocator_12962211300017
// MI455X (gfx1250) — compile-verified
//
#include <hip/hip_runtime.h>

#define K_LAYERS 50
#define NBATCH   2048
#define NUE      12
#define NRB      30
#define EPSF     1e-20f

// Native Clang vector type so __builtin_nontemporal_store accepts it
// (HIP's float4 is a struct and is rejected by the builtin).
typedef float v4f __attribute__((ext_vector_type(4)));

// Raw v_sqrt_f32 (~1 ulp) instead of the IEEE fixup sequence (~8 instr).
__device__ __forceinline__ float fsqrt_fast(float x) {
    return __builtin_amdgcn_sqrtf(x);
}

// v_rcp_f32 (~1 ulp) + one Newton-Raphson step: ~0.5-1 ulp in 3 instructions
// vs ~9 for the IEEE divide expansion. Denominators here are bounded into the
// normal range (>= ~1e-20 after the +eps guards), so no fixup is needed.
__device__ __forceinline__ float frcp_fast(float x) {
    float r = __builtin_amdgcn_rcpf(x);
    r = __builtin_fmaf(__builtin_fmaf(-x, r, 1.0f), r, r);  // r*(2 - x*r)
    return r;
}

// Pairwise tree sum of 12 values: 4-add critical path instead of 11.
__device__ __forceinline__ float tree_sum12(const float* v) {
    float s01 = v[0] + v[1],  s23 = v[2] + v[3];
    float s45 = v[4] + v[5],  s67 = v[6] + v[7];
    float s89 = v[8] + v[9],  sab = v[10] + v[11];
    float t0 = s01 + s23, t1 = s45 + s67, t2 = s89 + sab;
    return (t0 + t1) + t2;
}

// One thread owns one (batch, rb) pair: a 12-element (UE axis) problem kept
// entirely in VGPRs across all 50 layers. VALU-bound ~50x over HBM: the whole
// game is register residency + ILP. No matmul structure -> WMMA would only
// add layout traffic around an 11-add reduction.
__global__ __launch_bounds__(256) void ura_kernel(
    const float* __restrict__ a_init,
    const float* __restrict__ Hm,
    const float* __restrict__ Pm,
    const float* __restrict__ n0v,
    const float* __restrict__ etas,
    float* __restrict__ out)
{
    __shared__ float s_eta[K_LAYERS];

    // Stage the 50 per-layer step sizes into LDS via the CDNA5 async
    // global->LDS copy path (ASYNCcnt-tracked), then wait + block barrier.
    if (threadIdx.x < K_LAYERS) {
        unsigned lds_off = (unsigned)(size_t)(&s_eta[threadIdx.x]);
        unsigned long long gaddr = (unsigned long long)(etas + threadIdx.x);
        asm volatile("global_load_async_to_lds_b32 %0, %1, off"
                     :: "v"(lds_off), "v"(gaddr)
                     : "memory");
    }
    asm volatile("s_wait_asynccnt 0" ::: "memory");
    __syncthreads();

    const int tid = blockIdx.x * blockDim.x + threadIdx.x;
    const int NPAIR = NBATCH * NRB;          // 61440, exact multiple of 256
    if (tid >= NPAIR) return;
    const int b = tid / NRB;
    const int r = tid - b * NRB;

    const int base = b * (NUE * NRB) + r;    // element [b][u][r], stride NRB in u

    // gfx1250 prefetch (global_prefetch_b8) for the strided rows we pull next.
    __builtin_prefetch(a_init + base, 0, 1);
    __builtin_prefetch(Hm + base, 0, 1);
    __builtin_prefetch(Pm + base, 0, 1);

    float a[NUE], Hi[NUE], Pt[NUE], PH[NUE], sqPH[NUE];
    const float n0b = n0v[b];

    // Single-touch streaming inputs: non-temporal loads (TH_NT). Prologue math
    // stays IEEE (runs once; the loop below runs 50x).
#pragma unroll
    for (int u = 0; u < NUE; ++u) {
        const int idx = base + u * NRB;
        const float h = __builtin_nontemporal_load(Hm + idx);
        const float hi = 1.0f / h;           // channel gains bounded >= 0.1
        Hi[u]   = hi;
        a[u]    = __builtin_nontemporal_load(a_init + idx);
        Pt[u]   = __builtin_nontemporal_load(Pm + idx);
        PH[u]   = Pt[u] * hi;                // loop-invariant: Pt * (1/H)
        sqPH[u] = sqrtf(PH[u] + EPSF);       // loop-invariant: sqrt(Pt*Hi+eps)
    }

#pragma unroll 1
    for (int k = 0; k < K_LAYERS; ++k) {
        const float eta = s_eta[k];

        float aH[NUE];
#pragma unroll
        for (int u = 0; u < NUE; ++u) aH[u] = a[u] * Hi[u];
        const float sumAH = tree_sum12(aH);

        float A[NUE], g[NUE];
#pragma unroll
        for (int u = 0; u < NUE; ++u) {
            const float other  = sumAH - aH[u];
            const float interf = __builtin_fmaf(Pt[u], other, n0b);
            const float num    = fsqrt_fast(__builtin_fmaf(a[u], PH[u], EPSF)); // Q_vec
            const float w0     = num * frcp_fast(interf + EPSF);
            const float w      = (a[u] >= EPSF) ? w0 : 0.0f;
            const float wn     = w * num;
            const float ww     = w * w;
            // s = 2*w*num - w^2*interf ; analytically s >= 0 so denom >= 1 and
            // the reference's "+1e-20" is an exact f32 no-op.
            const float s      = __builtin_fmaf(2.0f, wn, -(ww * interf));
            const float denom  = 1.0f + s;
            const float sqa    = fsqrt_fast(a[u] + EPSF);
            const float rq     = frcp_fast(sqa * denom);   // 1/(sqrt(a+eps)*denom)
            // grad_direct = w*sqrt(Pt*Hi+eps) / (sqrt(a+eps)*denom)
            float gd           = w * sqPH[u] * rq;
            gd                 = (a[u] < EPSF) ? 0.0f : gd;
            // A = w^2 * Pt / denom  (1/denom == rq*sqa)
            const float Au     = ww * Pt[u] * (rq * sqa);
            A[u] = Au;
            g[u] = gd;
        }
        const float sumA = tree_sum12(A);

#pragma unroll
        for (int u = 0; u < NUE; ++u) {
            const float otherA = sumA - A[u];
            const float upd    = __builtin_fmaf(-Hi[u], otherA, g[u]); // gd + grad_interf
            const float an     = __builtin_fmaf(eta, upd, a[u]);
            a[u] = fminf(fmaxf(an, 0.0f), 1.0f);
        }
    }

    // Output [b][r][u]: 12 contiguous floats, 48B (16B aligned) per thread ->
    // three b128 non-temporal stores (output is never re-read).
    v4f* o = (v4f*)(out + ((size_t)(b * NRB + r)) * NUE);
    __builtin_nontemporal_store((v4f){a[0], a[1], a[2],  a[3]},  o + 0);
    __builtin_nontemporal_store((v4f){a[4], a[5], a[6],  a[7]},  o + 1);
    __builtin_nontemporal_store((v4f){a[8], a[9], a[10], a[11]}, o + 2);
}

extern "C" void kernel_launch(void* const* d_in, const int* in_sizes, int n_in,
                              void* d_out, int out_size, void* d_ws, size_t ws_size,
                              hipStream_t stream) {
    const float* a_init = (const float*)d_in[0];
    const float* H      = (const float*)d_in[1];
    const float* P      = (const float*)d_in[2];
    const float* n0     = (const float*)d_in[3];
    const float* etas   = (const float*)d_in[4];
    float* out          = (float*)d_out;

    const int npair = NBATCH * NRB;                  // 61440
    dim3 block(256);                                 // 8 wave32 waves / block
    dim3 grid((npair + block.x - 1) / block.x);      // 240 blocks
    ura_kernel<<<grid, block, 0, stream>>>(a_init, H, P, n0, etas, out);
}